// GaussianLSS_7533372637356
// MI455X (gfx1250) — compile-verified
//
#include <hip/hip_runtime.h>

typedef float v2f __attribute__((ext_vector_type(2)));
typedef float v8f __attribute__((ext_vector_type(8)));

#define GEMBED   128
#define GNG      432          // 6 cams * 6 * 12
#define GCHUNK   16           // gaussians per chunk
#define GNCHUNK  27           // 432 / 16
#define FPITCH   (GEMBED + 8) // 136: 2*136 mod 64 == 16 -> A-frag reads bank-disjoint
#define PIX_WAVE 16
#define WAVES_WG 8
#define PIX_WG   (PIX_WAVE * WAVES_WG)  // 128 pixels per workgroup
#define GTHRESH  0.01f

// ---------------------------------------------------------------------------
// Pre-pass: per-gaussian 2D params for all 4 scales + num_gaussians scalar.
// params[g*8] = { pr, pc, va, vb, vc, det, opac, 0 }, keep-mask folded in.
// ---------------------------------------------------------------------------
__global__ void gauss_prepass(const float* __restrict__ m0, const float* __restrict__ u0, const float* __restrict__ o0,
                              const float* __restrict__ m1, const float* __restrict__ u1, const float* __restrict__ o1,
                              const float* __restrict__ m2, const float* __restrict__ u2, const float* __restrict__ o2,
                              const float* __restrict__ m3, const float* __restrict__ u3, const float* __restrict__ o3,
                              float* __restrict__ params, float* __restrict__ numg)
{
    const int tid = threadIdx.x;
    if (tid == 0) *numg = 0.0f;
    __syncthreads();

    const float* Ms[4] = {m0, m1, m2, m3};
    const float* Us[4] = {u0, u1, u2, u3};
    const float* Os[4] = {o0, o1, o2, o3};
    const float  Hs[4] = {25.f, 50.f, 100.f, 200.f};

    float cnt = 0.0f;
    for (int g = tid; g < 4 * GNG; g += blockDim.x) {
        const int s = g / GNG;
        const int i = g - s * GNG;
        const float* m = Ms[s] + (size_t)i * 3;
        const float* u = Us[s] + (size_t)i * 9;
        float o = Os[s][i];
        const float H  = Hs[s];
        const float sh = H * 0.01f;           // sh == sw (square images)
        const float c00 = fmaxf(u[0], 1.0f);  // c9[0]
        const float c01 = fmaxf(u[1], 1.0f);  // c9[1]
        const float c11 = fmaxf(u[4], 1.0f);  // c9[4]
        const float va  = sh * sh * c11 + 0.3f;
        const float vb  = sh * sh * c01;
        const float vc  = sh * sh * c00 + 0.3f;
        float det = va * vc - vb * vb;
        const float pr = -sh * m[1] + H * 0.5f;
        const float pc = -sh * m[0] + H * 0.5f;
        if (o > GTHRESH) cnt += 1.0f;
        const bool keep = (o > GTHRESH) && (det > 0.0f);
        if (!keep) { o = 0.0f; det = 1.0f; }  // alpha forced to 0 downstream
        float* Pp = params + (size_t)g * 8;
        Pp[0] = pr; Pp[1] = pc; Pp[2] = va; Pp[3] = vb;
        Pp[4] = vc; Pp[5] = det; Pp[6] = o;  Pp[7] = 0.0f;
    }
    atomicAdd(numg, cnt);
}

// ---------------------------------------------------------------------------
// Fused render: all 4 scales in one grid. 8 waves/WG, 16 pixels/wave.
// Weights scanned sequentially per pixel (cumulative transmittance), GEMM
// out[d,p] += F[n,d] * w[n,p] done with v_wmma_f32_16x16x4_f32.
// ---------------------------------------------------------------------------
struct RenderArgs {
    const float* feat[4];
    const float* params;       // base of packed params (4 * 432 * 8 floats)
    float* out;                // base of concatenated outputs
    int bstart[4];             // first block of each scale
    int H[4];
    unsigned long long ooff[4];
};

__global__ __launch_bounds__(256) void gauss_render(RenderArgs A)
{
    __shared__ float lds_f[GCHUNK][FPITCH];              // feature chunk (16 x 128, padded)
    __shared__ float lds_w[WAVES_WG][GCHUNK][PIX_WAVE];  // per-wave weight tile
    __shared__ float lds_p[GCHUNK][8];                   // gaussian params chunk

    const int bs = blockIdx.x;
    int s = 3;
    if      (bs < A.bstart[1]) s = 0;
    else if (bs < A.bstart[2]) s = 1;
    else if (bs < A.bstart[3]) s = 2;
    const int blk = bs - A.bstart[s];

    const float* __restrict__ feat   = A.feat[s];
    const float* __restrict__ params = A.params + (size_t)s * GNG * 8;
    float* __restrict__ out          = A.out + A.ooff[s];
    const int H = A.H[s];
    const int W = H;
    const int P = H * W;

    const int tid  = threadIdx.x;
    const int wave = tid >> 5;
    const int lane = tid & 31;
    const int l16  = lane & 15;
    const int hi   = lane >> 4;              // lane half (0/1)
    const int pbase = blk * PIX_WG + wave * PIX_WAVE;

    // Pixel owned by this lane during the weight scan (lanes 0..15 only).
    const int p   = pbase + lane;
    const int pcl = (p < P) ? p : (P - 1);
    const int row = pcl / W;
    const float rf = (float)row;
    const float cf = (float)(pcl - row * W);
    float T = 1.0f;                          // transmittance, persists across chunks

    v8f acc[8];
    #pragma unroll
    for (int t = 0; t < 8; ++t) {
        #pragma unroll
        for (int i = 0; i < 8; ++i) acc[t][i] = 0.0f;
    }

    for (int ch = 0; ch < GNCHUNK; ++ch) {
        const int nbase = ch * GCHUNK;
        __syncthreads();   // protect LDS reuse from previous iteration's readers

        // Stage feature chunk: lds_f[kk][d] = F[nbase+kk][d]; consecutive tids
        // walk consecutive gaussians (coalesced along hw in global memory).
        #pragma unroll
        for (int e = tid; e < GCHUNK * GEMBED; e += 256) {
            const int kk  = e & (GCHUNK - 1);
            const int d   = e >> 4;
            const int n   = nbase + kk;
            const int cam = n / 72;
            const int hw  = n - cam * 72;
            lds_f[kk][d] = feat[((size_t)cam * GEMBED + d) * 72 + hw];
            if (ch + 1 < GNCHUNK && (e & 15) == 0) {
                const int n2   = n + GCHUNK;
                const int cam2 = n2 / 72;
                const int hw2  = n2 - cam2 * 72;
                __builtin_prefetch(&feat[((size_t)cam2 * GEMBED + d) * 72 + hw2], 0, 1);
            }
        }
        // Stage params chunk (16 gaussians x 8 floats).
        if (tid < GCHUNK * 8) {
            lds_p[tid >> 3][tid & 7] = params[(size_t)(nbase + (tid >> 3)) * 8 + (tid & 7)];
        }
        __syncthreads();

        // Sequential compositing scan: one pixel per lane (lanes 0..15).
        if (lane < 16) {
            #pragma unroll
            for (int kk = 0; kk < GCHUNK; ++kk) {
                const float pr  = lds_p[kk][0];
                const float pcn = lds_p[kk][1];
                const float va  = lds_p[kk][2];
                const float vb  = lds_p[kk][3];
                const float vc  = lds_p[kk][4];
                const float det = lds_p[kk][5];
                const float o   = lds_p[kk][6];
                const float dr  = pr - rf;
                const float dc  = pcn - cf;
                const float num = vb * dr * dc - 0.5f * (vc * dr * dr + va * dc * dc);
                const float pw  = num / det;
                const float al  = fminf(0.99f, o * __expf(fminf(pw, 0.0f)));
                const float a2  = ((pw <= 0.0f) && (al >= (1.0f / 255.0f))) ? al : 0.0f;
                lds_w[wave][kk][lane] = a2 * T;
                T *= (1.0f - a2);
            }
        }
        __syncthreads();

        // GEMM phase: 4 K-steps of 4, 8 channel tiles. EXEC is all-1s here.
        #pragma unroll
        for (int j = 0; j < GCHUNK / 4; ++j) {
            const int k0 = j * 4 + 2 * hi;   // A/B VGPR0 holds K=k0 (lo half) / k0+2 (hi half)
            v2f bf;
            bf.x = lds_w[wave][k0    ][l16];
            bf.y = lds_w[wave][k0 + 1][l16];
            #pragma unroll
            for (int t = 0; t < 8; ++t) {
                v2f af;
                af.x = lds_f[k0    ][t * 16 + l16];
                af.y = lds_f[k0 + 1][t * 16 + l16];
                acc[t] = __builtin_amdgcn_wmma_f32_16x16x4_f32(
                    /*neg_a=*/false, af, /*neg_b=*/false, bf,
                    /*c_mod=*/(short)0, acc[t], /*reuse_a=*/false, /*reuse_b=*/false);
            }
        }
    }

    // Store: 16x16 f32 C/D layout -> VGPR i is M=i (lanes 0-15) / M=i+8 (16-31).
    const int pout = pbase + l16;
    if (pout < P) {
        #pragma unroll
        for (int t = 0; t < 8; ++t) {
            const int dbase = t * 16 + hi * 8;
            #pragma unroll
            for (int i = 0; i < 8; ++i) {
                out[(size_t)(dbase + i) * P + pout] = acc[t][i];
            }
        }
    }
}

// ---------------------------------------------------------------------------
extern "C" void kernel_launch(void* const* d_in, const int* in_sizes, int n_in,
                              void* d_out, int out_size, void* d_ws, size_t ws_size,
                              hipStream_t stream)
{
    (void)in_sizes; (void)n_in; (void)ws_size;
    // setup_inputs() dict order: per scale s: features, means, uncert, opac; then b, n.
    const float* feats[4]; const float* means[4]; const float* uncs[4]; const float* ops[4];
    for (int s = 0; s < 4; ++s) {
        feats[s] = (const float*)d_in[4 * s + 0];
        means[s] = (const float*)d_in[4 * s + 1];
        uncs[s]  = (const float*)d_in[4 * s + 2];
        ops[s]   = (const float*)d_in[4 * s + 3];
    }
    float* out    = (float*)d_out;
    float* params = (float*)d_ws;               // 4 * 432 * 8 floats = 55 KB
    float* numg   = out + (out_size - 1);       // last output element = num_gaussians

    hipLaunchKernelGGL(gauss_prepass, dim3(1), dim3(256), 0, stream,
                       means[0], uncs[0], ops[0], means[1], uncs[1], ops[1],
                       means[2], uncs[2], ops[2], means[3], uncs[3], ops[3],
                       params, numg);

    RenderArgs A;
    A.params = params;
    A.out    = out;
    const int Hs[4] = {25, 50, 100, 200};
    int bcum = 0;
    unsigned long long ocum = 0;
    for (int s = 0; s < 4; ++s) {
        A.feat[s]   = feats[s];
        A.H[s]      = Hs[s];
        A.bstart[s] = bcum;
        A.ooff[s]   = ocum;
        const int Pn = Hs[s] * Hs[s];
        bcum += (Pn + PIX_WG - 1) / PIX_WG;
        ocum += (unsigned long long)GEMBED * Pn;
    }
    hipLaunchKernelGGL(gauss_render, dim3(bcum), dim3(256), 0, stream, A);
}